// GraspNetWithAttention_29446295781560
// MI455X (gfx1250) — compile-verified
//
#include <hip/hip_runtime.h>
#include <hip/hip_bf16.h>
#include <math.h>
#include <stdint.h>

#define MSEQ 2048
#define NB   8

typedef __attribute__((ext_vector_type(2))) float v2f;
typedef __attribute__((ext_vector_type(4))) float v4f;
typedef __attribute__((ext_vector_type(8))) float v8f;

__device__ __forceinline__ v8f wmma_f32_k4(v2f a, v2f b, v8f c) {
  // D(16x16,f32) = A(16x4,f32) * B(4x16,f32) + C
  return __builtin_amdgcn_wmma_f32_16x16x4_f32(false, a, false, b, (short)0, c,
                                               false, false);
}

// ---- optional gfx1250 async global->LDS path (guarded; sync fallback) ------
#if defined(__has_builtin)
#if __has_builtin(__builtin_amdgcn_global_load_async_to_lds_b32) && \
    __has_builtin(__builtin_amdgcn_s_wait_asynccnt)
#define USE_ASYNC_LDS 1
#endif
#endif

#ifdef USE_ASYNC_LDS
typedef __attribute__((address_space(1))) int* gas_i32p;
typedef __attribute__((address_space(3))) int* las_i32p;
__device__ __forceinline__ void async_copy_b32(const float* gptr, float* lptr) {
  __builtin_amdgcn_global_load_async_to_lds_b32(
      (gas_i32p)(uintptr_t)gptr,
      (las_i32p)(uintptr_t)(uint32_t)(uintptr_t)lptr,
      0, 0);
}
#endif

// ---------------------------------------------------------------------------
// feat = concat(g@Wg+bg, c@Wc+bc, p@Wp+bp)  -> [B*M, 48]
// ---------------------------------------------------------------------------
__global__ void k_feat(const float* __restrict__ g, const float* __restrict__ cc,
                       const float* __restrict__ p,
                       const float* __restrict__ Wg, const float* __restrict__ bg,
                       const float* __restrict__ Wc, const float* __restrict__ bc,
                       const float* __restrict__ Wp, const float* __restrict__ bp,
                       float* __restrict__ feat) {
  int idx = blockIdx.x * blockDim.x + threadIdx.x;     // B*M*48 threads
  int row = idx / 48, col = idx % 48;
  float acc;
  if (col < 16) {
    acc = bg[col];
    #pragma unroll
    for (int i = 0; i < 10; ++i) acc += g[row * 10 + i] * Wg[i * 16 + col];
  } else if (col < 32) {
    int c0 = col - 16;
    acc = bc[c0];
    #pragma unroll
    for (int i = 0; i < 4; ++i) acc += cc[row * 4 + i] * Wc[i * 16 + c0];
  } else {
    int c0 = col - 32;
    acc = bp[c0];
    #pragma unroll
    for (int i = 0; i < 8; ++i) acc += p[row * 8 + i] * Wp[i * 16 + c0];
  }
  feat[idx] = acc;
}

// ---------------------------------------------------------------------------
// Generic fp32 WMMA GEMM: C[R,N] = A[R,K] @ W[K,N] + bias[N]  (optional relu)
// One wave (32 threads) per 16x16 output tile. K % 4 == 0.
// ---------------------------------------------------------------------------
__global__ void k_gemm(const float* __restrict__ A, const float* __restrict__ W,
                       const float* __restrict__ bias, float* __restrict__ C,
                       int K, int N, int relu) {
  int lane = threadIdx.x;
  int half = lane >> 4;
  int n = lane & 15, m = n;
  int col0 = blockIdx.x * 16;
  int row0 = blockIdx.y * 16;
  float bv = bias[col0 + n];
  v8f c;
  #pragma unroll
  for (int v = 0; v < 8; ++v) c[v] = bv;
  for (int k0 = 0; k0 < K; k0 += 4) {
    int kk = k0 + 2 * half;
    v2f a, b;
    const float* ap = A + (size_t)(row0 + m) * K + kk;
    a.x = ap[0];
    a.y = ap[1];
    b.x = W[(size_t)kk * N + col0 + n];
    b.y = W[(size_t)(kk + 1) * N + col0 + n];
    c = wmma_f32_k4(a, b, c);
  }
  #pragma unroll
  for (int v = 0; v < 8; ++v) {
    float o = c[v];
    if (relu) o = fmaxf(o, 0.f);
    C[(size_t)(row0 + v + 8 * half) * N + col0 + n] = o;
  }
}

// ---------------------------------------------------------------------------
// MHA flash attention (H=4, HD=12). qkv layout [B,M,144]: q|k|v blocks of 48,
// head h at offset h*12 within each block. Writes o -> obuf [B,M,48].
// Block = 256 threads = 8 waves; each wave owns 16 query rows.
// ---------------------------------------------------------------------------
__global__ void k_mha(const float* __restrict__ qkv, float* __restrict__ obuf) {
  __shared__ float Kt[12 * 64];        // K^T chunk: [d][key]
  __shared__ float Vc[64 * 16];        // V chunk:   [key][d] (cols 12..15 = 0)
  __shared__ float Pscr[8 * 16 * 17];  // per-wave P staging (C-layout -> A-layout)
  int tid = threadIdx.x;
  int wid = tid >> 5, lane = tid & 31;
  int half = lane >> 4;
  int n = lane & 15, m = n;
  int b = blockIdx.z, h = blockIdx.y;
  int q0 = blockIdx.x * 128 + wid * 16;
  const float sc = rsqrtf(12.f);

  // zero V pad columns once (cols 12..15 are never written by the chunk loads
  // on the async path)
  for (int idx = tid; idx < 64 * 16; idx += 256) Vc[idx] = 0.f;

  v2f aq[3];
  {
    const float* qp = qkv + (size_t)(b * MSEQ + q0 + m) * 144 + h * 12;
    #pragma unroll
    for (int j = 0; j < 3; ++j) {
      int kk = 4 * j + 2 * half;
      aq[j].x = qp[kk] * sc;
      aq[j].y = qp[kk + 1] * sc;
    }
  }

  v8f O;
  float mcur[8], lcur[8];
  #pragma unroll
  for (int v = 0; v < 8; ++v) { O[v] = 0.f; mcur[v] = -1e30f; lcur[v] = 0.f; }
  float* Pw = Pscr + wid * 16 * 17;

  for (int nc0 = 0; nc0 < MSEQ; nc0 += 64) {
    __syncthreads();
#ifdef USE_ASYNC_LDS
    for (int idx = tid; idx < 64 * 16; idx += 256) {
      int kl = idx >> 4, d = idx & 15;
      const float* kp = qkv + (size_t)(b * MSEQ + nc0 + kl) * 144 + h * 12;
      if (d < 12) {
        async_copy_b32(kp + 48 + d, &Kt[d * 64 + kl]);
        async_copy_b32(kp + 96 + d, &Vc[kl * 16 + d]);
      }
    }
    __builtin_amdgcn_s_wait_asynccnt(0);
#else
    for (int idx = tid; idx < 64 * 16; idx += 256) {
      int kl = idx >> 4, d = idx & 15;
      const float* kp = qkv + (size_t)(b * MSEQ + nc0 + kl) * 144 + h * 12;
      if (d < 12) {
        Kt[d * 64 + kl] = kp[48 + d];
        Vc[kl * 16 + d] = kp[96 + d];
      }
    }
#endif
    __syncthreads();
    for (int s = 0; s < 4; ++s) {
      v8f cS;
      #pragma unroll
      for (int v = 0; v < 8; ++v) cS[v] = 0.f;
      #pragma unroll
      for (int j = 0; j < 3; ++j) {
        int kk = 4 * j + 2 * half;
        v2f bb;
        bb.x = Kt[kk * 64 + s * 16 + n];
        bb.y = Kt[(kk + 1) * 64 + s * 16 + n];
        cS = wmma_f32_k4(aq[j], bb, cS);
      }
      #pragma unroll
      for (int v = 0; v < 8; ++v) {
        float sv = cS[v];
        float mx = sv;
        #pragma unroll
        for (int off = 8; off >= 1; off >>= 1)
          mx = fmaxf(mx, __shfl_xor(mx, off, 16));
        float mn = fmaxf(mcur[v], mx);
        float rescale = __expf(mcur[v] - mn);
        float pv = __expf(sv - mn);
        float ps = pv;
        #pragma unroll
        for (int off = 8; off >= 1; off >>= 1) ps += __shfl_xor(ps, off, 16);
        lcur[v] = lcur[v] * rescale + ps;
        mcur[v] = mn;
        O[v] *= rescale;
        Pw[(v + 8 * half) * 17 + n] = pv;
      }
      #pragma unroll
      for (int j = 0; j < 4; ++j) {
        int kk = 4 * j + 2 * half;
        v2f ap, bv2;
        ap.x = Pw[m * 17 + kk];
        ap.y = Pw[m * 17 + kk + 1];
        bv2.x = Vc[(s * 16 + kk) * 16 + n];
        bv2.y = Vc[(s * 16 + kk + 1) * 16 + n];
        O = wmma_f32_k4(ap, bv2, O);
      }
    }
  }
  #pragma unroll
  for (int v = 0; v < 8; ++v) {
    if (n < 12)
      obuf[(size_t)(b * MSEQ + q0 + v + 8 * half) * 48 + h * 12 + n] =
          O[v] / lcur[v];
  }
}

// ---------------------------------------------------------------------------
// Second attention with additive relation bias. qkv2 [B,M,48] = Q|K|V (16 ea).
// bias[m,n] = r[b,m,n,:]·Wr + br, fused while streaming r once (536 MB pass,
// non-temporal: zero reuse, keep it out of WGP$/L2).
// Writes z rows -> zbuf [B,M,16].
// ---------------------------------------------------------------------------
__global__ void k_attn2(const float* __restrict__ qkv2, const float* __restrict__ r,
                        const float* __restrict__ Wr, const float* __restrict__ br,
                        float* __restrict__ zbuf) {
  __shared__ float Kt[16 * 64];
  __shared__ float Vc[64 * 16];
  __shared__ float Pscr[8 * 16 * 17];
  int tid = threadIdx.x;
  int wid = tid >> 5, lane = tid & 31;
  int half = lane >> 4;
  int n = lane & 15, m = n;
  int b = blockIdx.z;
  int q0 = blockIdx.x * 128 + wid * 16;
  float w0 = Wr[0], w1 = Wr[1], w2 = Wr[2], w3 = Wr[3], br0 = br[0];

  v2f aq[4];
  {
    const float* qp = qkv2 + (size_t)(b * MSEQ + q0 + m) * 48;
    #pragma unroll
    for (int j = 0; j < 4; ++j) {
      int kk = 4 * j + 2 * half;
      aq[j].x = qp[kk] * 0.25f;   // 1/sqrt(16)
      aq[j].y = qp[kk + 1] * 0.25f;
    }
  }

  v8f O;
  float mcur[8], lcur[8];
  #pragma unroll
  for (int v = 0; v < 8; ++v) { O[v] = 0.f; mcur[v] = -1e30f; lcur[v] = 0.f; }
  float* Pw = Pscr + wid * 16 * 17;

  for (int nc0 = 0; nc0 < MSEQ; nc0 += 64) {
    __syncthreads();
#ifdef USE_ASYNC_LDS
    for (int idx = tid; idx < 64 * 16; idx += 256) {
      int kl = idx >> 4, d = idx & 15;
      const float* kp = qkv2 + (size_t)(b * MSEQ + nc0 + kl) * 48;
      async_copy_b32(kp + 16 + d, &Kt[d * 64 + kl]);
      async_copy_b32(kp + 32 + d, &Vc[kl * 16 + d]);
    }
    __builtin_amdgcn_s_wait_asynccnt(0);
#else
    for (int idx = tid; idx < 64 * 16; idx += 256) {
      int kl = idx >> 4, d = idx & 15;
      const float* kp = qkv2 + (size_t)(b * MSEQ + nc0 + kl) * 48;
      Kt[d * 64 + kl] = kp[16 + d];
      Vc[kl * 16 + d] = kp[32 + d];
    }
#endif
    __syncthreads();
    if (nc0 + 64 < MSEQ)
      __builtin_prefetch(
          r + ((size_t)(b * MSEQ + q0 + m) * MSEQ + nc0 + 64 + n) * 4, 0, 1);
    for (int s = 0; s < 4; ++s) {
      v8f cS;
      #pragma unroll
      for (int v = 0; v < 8; ++v) cS[v] = 0.f;
      #pragma unroll
      for (int j = 0; j < 4; ++j) {
        int kk = 4 * j + 2 * half;
        v2f bb;
        bb.x = Kt[kk * 64 + s * 16 + n];
        bb.y = Kt[(kk + 1) * 64 + s * 16 + n];
        cS = wmma_f32_k4(aq[j], bb, cS);
      }
      int ncol = nc0 + s * 16 + n;
      #pragma unroll
      for (int v = 0; v < 8; ++v) {
        // single-use 536 MB stream: non-temporal 128-bit load
        const v4f rv = __builtin_nontemporal_load((const v4f*)(
            r + ((size_t)(b * MSEQ + q0 + v + 8 * half) * MSEQ + ncol) * 4));
        cS[v] += rv.x * w0 + rv.y * w1 + rv.z * w2 + rv.w * w3 + br0;
      }
      #pragma unroll
      for (int v = 0; v < 8; ++v) {
        float sv = cS[v];
        float mx = sv;
        #pragma unroll
        for (int off = 8; off >= 1; off >>= 1)
          mx = fmaxf(mx, __shfl_xor(mx, off, 16));
        float mn = fmaxf(mcur[v], mx);
        float rescale = __expf(mcur[v] - mn);
        float pv = __expf(sv - mn);
        float ps = pv;
        #pragma unroll
        for (int off = 8; off >= 1; off >>= 1) ps += __shfl_xor(ps, off, 16);
        lcur[v] = lcur[v] * rescale + ps;
        mcur[v] = mn;
        O[v] *= rescale;
        Pw[(v + 8 * half) * 17 + n] = pv;
      }
      #pragma unroll
      for (int j = 0; j < 4; ++j) {
        int kk = 4 * j + 2 * half;
        v2f ap, bv2;
        ap.x = Pw[m * 17 + kk];
        ap.y = Pw[m * 17 + kk + 1];
        bv2.x = Vc[(s * 16 + kk) * 16 + n];
        bv2.y = Vc[(s * 16 + kk + 1) * 16 + n];
        O = wmma_f32_k4(ap, bv2, O);
      }
    }
  }
  #pragma unroll
  for (int v = 0; v < 8; ++v)
    zbuf[(size_t)(b * MSEQ + q0 + v + 8 * half) * 16 + n] = O[v] / lcur[v];
}

// ---------------------------------------------------------------------------
// z_global[b,16] = mean_m zbuf[b,m,16]
// ---------------------------------------------------------------------------
__global__ void k_zmean(const float* __restrict__ zbuf, float* __restrict__ zg) {
  int b = blockIdx.x, d = threadIdx.x;  // 16 threads
  float s = 0.f;
  for (int mr = 0; mr < MSEQ; ++mr) s += zbuf[(size_t)(b * MSEQ + mr) * 16 + d];
  zg[b * 16 + d] = s / (float)MSEQ;
}

// zb[b,64] = z_global[b,:16] @ Ws1[48:64, :64]
__global__ void k_zb(const float* __restrict__ zg, const float* __restrict__ Ws1,
                     float* __restrict__ zb) {
  int b = blockIdx.x, j = threadIdx.x;  // 64 threads
  float s = 0.f;
  #pragma unroll
  for (int d = 0; d < 16; ++d) s += zg[b * 16 + d] * Ws1[(48 + d) * 64 + j];
  zb[b * 64 + j] = s;
}

// ---------------------------------------------------------------------------
// Heads: pos_pred = t1(relu'd) @ Wp2 + bp2 ; score = sigmoid(relu(t2+zb)@Ws2+bs2)
// ---------------------------------------------------------------------------
__global__ void k_heads(const float* __restrict__ t1, const float* __restrict__ t2,
                        const float* __restrict__ zb,
                        const float* __restrict__ Wp2, const float* __restrict__ bp2,
                        const float* __restrict__ Ws2, const float* __restrict__ bs2,
                        float* __restrict__ out) {
  int row = blockIdx.x * blockDim.x + threadIdx.x;  // B*M rows
  int b = row >> 11;
  const float* t1r = t1 + (size_t)row * 64;
  #pragma unroll
  for (int oc = 0; oc < 6; ++oc) {
    float acc = bp2[oc];
    for (int j = 0; j < 64; ++j) acc += t1r[j] * Wp2[j * 6 + oc];
    out[(size_t)row * 6 + oc] = acc;
  }
  float s = bs2[0];
  const float* t2r = t2 + (size_t)row * 64;
  const float* zr = zb + b * 64;
  for (int j = 0; j < 64; ++j) {
    float hj = fmaxf(t2r[j] + zr[j], 0.f);
    s += hj * Ws2[j];
  }
  out[(size_t)NB * MSEQ * 6 + row] = 1.f / (1.f + __expf(-s));
}

// ---------------------------------------------------------------------------
extern "C" void kernel_launch(void* const* d_in, const int* in_sizes, int n_in,
                              void* d_out, int out_size, void* d_ws, size_t ws_size,
                              hipStream_t stream) {
  const float* g    = (const float*)d_in[0];
  const float* c    = (const float*)d_in[1];
  const float* p    = (const float*)d_in[2];
  const float* r    = (const float*)d_in[3];
  const float* Wg   = (const float*)d_in[4];
  const float* bg   = (const float*)d_in[5];
  const float* Wc   = (const float*)d_in[6];
  const float* bc   = (const float*)d_in[7];
  const float* Wp   = (const float*)d_in[8];
  const float* bp   = (const float*)d_in[9];
  const float* Wi   = (const float*)d_in[10];
  const float* bi   = (const float*)d_in[11];
  const float* Wo   = (const float*)d_in[12];
  const float* bo   = (const float*)d_in[13];
  const float* Wr   = (const float*)d_in[14];
  const float* br   = (const float*)d_in[15];
  const float* Wqkv = (const float*)d_in[16];
  const float* bqkv = (const float*)d_in[17];
  const float* Wp1  = (const float*)d_in[18];
  const float* bp1  = (const float*)d_in[19];
  const float* Wp2  = (const float*)d_in[20];
  const float* bp2  = (const float*)d_in[21];
  const float* Ws1  = (const float*)d_in[22];
  const float* bs1  = (const float*)d_in[23];
  const float* Ws2  = (const float*)d_in[24];
  const float* bs2  = (const float*)d_in[25];

  float* ws = (float*)d_ws;
  const size_t RM = (size_t)NB * MSEQ;           // 16384 rows
  float* feat   = ws;                            // [RM,48]  (dead after qkv1)
  float* obuf   = ws;                            // reuses feat region
  float* qkv1   = ws + RM * 48;                  // [RM,144] (dead after MHA)
  float* fprime = ws + RM * 48 + RM * 144;       // [RM,48]
  float* qkv2   = fprime + RM * 48;              // [RM,48]
  float* zbuf   = qkv2 + RM * 48;                // [RM,16]
  float* zg     = zbuf + RM * 16;                // [8,16]
  float* zbb    = zg + NB * 16;                  // [8,64]
  float* t1     = qkv1;                          // [RM,64] reuses qkv1 region
  float* t2     = qkv1 + RM * 64;                // [RM,64]

  // 1) input projections
  k_feat<<<(RM * 48) / 256, 256, 0, stream>>>(g, c, p, Wg, bg, Wc, bc, Wp, bp, feat);
  // 2) MHA in-projection: [16384x48] @ [48x144]
  k_gemm<<<dim3(144 / 16, RM / 16), 32, 0, stream>>>(feat, Wi, bi, qkv1, 48, 144, 0);
  // 3) MHA flash attention
  k_mha<<<dim3(MSEQ / 128, 4, NB), 256, 0, stream>>>(qkv1, obuf);
  // 4) out-projection -> f'
  k_gemm<<<dim3(48 / 16, RM / 16), 32, 0, stream>>>(obuf, Wo, bo, fprime, 48, 48, 0);
  // 5) second qkv projection
  k_gemm<<<dim3(48 / 16, RM / 16), 32, 0, stream>>>(fprime, Wqkv, bqkv, qkv2, 48, 48, 0);
  // 6) MLP first layers (t1 relu'd; t2 pre-relu, z part added in heads)
  k_gemm<<<dim3(64 / 16, RM / 16), 32, 0, stream>>>(fprime, Wp1, bp1, t1, 48, 64, 1);
  k_gemm<<<dim3(64 / 16, RM / 16), 32, 0, stream>>>(fprime, Ws1, bs1, t2, 48, 64, 0);
  // 7) bias attention (streams r once, non-temporal)
  k_attn2<<<dim3(MSEQ / 128, 1, NB), 256, 0, stream>>>(qkv2, r, Wr, br, zbuf);
  // 8) global pooling + z @ Ws1_bottom
  k_zmean<<<NB, 16, 0, stream>>>(zbuf, zg);
  k_zb<<<NB, 64, 0, stream>>>(zg, Ws1, zbb);
  // 9) heads
  k_heads<<<RM / 128, 128, 0, stream>>>(t1, t2, zbb, Wp2, bp2, Ws2, bs2, (float*)d_out);
}